// SequenceGeneratorModel_63591285784709
// MI455X (gfx1250) — compile-verified
//
#include <hip/hip_runtime.h>
#include <math.h>

// ---------------------------------------------------------------------------
// CDNA5 (gfx1250) wave32 WMMA helpers: V_WMMA_F32_16X16X4_F32
//   A: 16x4 f32  -> 2 VGPRs/lane (lanes 0-15: K=k,k+1 ; lanes 16-31: K=k+2,k+3)
//   B: 4x16 f32  -> 2 VGPRs/lane (lanes 0-15: K=k,k+1 ; lanes 16-31: K=k+2,k+3; N=lane%16)
//   C/D: 16x16   -> 8 VGPRs (VGPR r: lanes 0-15 M=r, lanes 16-31 M=r+8; N=lane%16)
// ---------------------------------------------------------------------------
typedef __attribute__((ext_vector_type(2))) float v2f;
typedef __attribute__((ext_vector_type(8))) float v8f;

__device__ __forceinline__ v8f wmma_f32_16x16x4(v2f a, v2f b, v8f c) {
  // 8 args: (neg_a, A, neg_b, B, c_mod, C, reuse_a, reuse_b)
  return __builtin_amdgcn_wmma_f32_16x16x4_f32(false, a, false, b, (short)0, c,
                                               false, false);
}

// ---------------------------------------------------------------------------
// Generic M=64 GEMM:  C[64xN] = A[64xK] @ B[KxN] + bias[N]
// One wave owns a 64x16 output strip (4 WMMA accumulators), so every B (weight)
// element is read exactly once -> the logits GEMM streams Wo at exactly 200 MB,
// the HBM roofline floor (~8.6 us at 23.3 TB/s). K,N are template constants so
// unrolled B fetches fold into 24-bit immediate offsets (no per-step v_mul_u64,
// no address-register serialization): one clause of 16 NT loads per unroll-8
// block keeps ~2KB/wave of the Wo stream in flight. The 8 waves of a workgroup
// walk the same 1KB A-slab per K-step, so A stays WGP$-resident.
// NT_STORE: true for logits (write-once, never re-read), false for activations.
// Requires: K % 4 == 0, N % 16 == 0 (50000 = 3125*16).
// ---------------------------------------------------------------------------
template <int K, int N, bool NT_STORE>
__global__ __launch_bounds__(256) void gemm64_wmma(
    const float* __restrict__ A, const float* __restrict__ B,
    const float* __restrict__ bias, float* __restrict__ C) {
  const int gwave = (blockIdx.x * 256 + threadIdx.x) >> 5;
  const int lane  = threadIdx.x & 31;
  const int n0    = gwave * 16;
  if (n0 >= N) return;              // wave-uniform: EXEC stays all-ones for WMMA
  const int half = lane >> 4;       // 0: lanes 0-15, 1: lanes 16-31
  const int l16  = lane & 15;
  const int aOff = half * 2;        // upper half-wave handles K+2,K+3

  const float* a0 = A + (size_t)(0  + l16) * K + aOff;
  const float* a1 = A + (size_t)(16 + l16) * K + aOff;
  const float* a2 = A + (size_t)(32 + l16) * K + aOff;
  const float* a3 = A + (size_t)(48 + l16) * K + aOff;
  const float* bp = B + (size_t)aOff * N + n0 + l16;   // lane's B column base

  v8f c0 = {}, c1 = {}, c2 = {}, c3 = {};
#pragma unroll 8
  for (int k = 0; k < K; k += 4) {
    v2f af0 = *(const v2f*)(a0 + k);
    v2f af1 = *(const v2f*)(a1 + k);
    v2f af2 = *(const v2f*)(a2 + k);
    v2f af3 = *(const v2f*)(a3 + k);
    v2f bf;
    bf.x = __builtin_nontemporal_load(bp + (size_t)(k + 0) * N);  // imm offset
    bf.y = __builtin_nontemporal_load(bp + (size_t)(k + 1) * N);  // imm offset
    c0 = wmma_f32_16x16x4(af0, bf, c0);
    c1 = wmma_f32_16x16x4(af1, bf, c1);
    c2 = wmma_f32_16x16x4(af2, bf, c2);
    c3 = wmma_f32_16x16x4(af3, bf, c3);
  }

  const float  bv  = bias[n0 + l16];
  const size_t col = (size_t)n0 + l16;
#pragma unroll
  for (int r = 0; r < 8; ++r) {
    const int row = r + half * 8;
    float* p0 = &C[(size_t)(row +  0) * N + col];
    float* p1 = &C[(size_t)(row + 16) * N + col];
    float* p2 = &C[(size_t)(row + 32) * N + col];
    float* p3 = &C[(size_t)(row + 48) * N + col];
    if (NT_STORE) {
      __builtin_nontemporal_store(c0[r] + bv, p0);
      __builtin_nontemporal_store(c1[r] + bv, p1);
      __builtin_nontemporal_store(c2[r] + bv, p2);
      __builtin_nontemporal_store(c3[r] + bv, p3);
    } else {
      *p0 = c0[r] + bv;
      *p1 = c1[r] + bv;
      *p2 = c2[r] + bv;
      *p3 = c3[r] + bv;
    }
  }
}

// ---------------------------------------------------------------------------
// Fused Bahdanau attention: one workgroup (4 waves) per batch element b.
// Wave w owns s-rows [16w, 16w+16). All 4 waves consume IDENTICAL W1 B-tiles,
// so each 256x16 tile is staged once per workgroup in LDS (cuts W1's L2
// traffic 4x: 128MB -> 32MB). LDS layout is transposed [n][k] with a 260-float
// pitch: the 32 lanes of a B-fragment ds_load_b64 hit banks (4n+k+{0..3})%64
// == each of the 64 banks exactly once -> conflict-free.
// tanh(+b1+hid_proj)*V is reduced per lane, then a deterministic shuffle tree
// collapses the 16 lanes sharing a row. Softmax + context in LDS.
// ---------------------------------------------------------------------------
#define W1_PITCH 260
__global__ __launch_bounds__(128) void attention_kernel(
    const float* __restrict__ qs,   // 64 x 64 x 256
    const float* __restrict__ W1,   // 256 x 512
    const float* __restrict__ b1,   // 512
    const float* __restrict__ hp,   // 64 x 512 = hidden@W2 + b2
    const float* __restrict__ Vw,   // 512
    const float* __restrict__ bV,   // 1
    float* __restrict__ wout,       // 64 x 64 attention weights (output)
    float* __restrict__ ctx)        // 64 x 256 context (workspace)
{
  const int b    = blockIdx.x;
  const int tid  = threadIdx.x;
  const int wave = tid >> 5;
  const int lane = tid & 31;
  const int half = lane >> 4;
  const int l16  = lane & 15;
  const int aOff = half * 2;

  __shared__ float s_score[64];
  __shared__ float s_red[64];
  __shared__ float s_w[64];
  __shared__ float ldsW[16][W1_PITCH];   // W1 tile, transposed [n][k]

  const float* Aq   = qs + (size_t)b * 64 * 256;
  const float* arow = Aq + (size_t)(wave * 16 + l16) * 256 + aOff;
  const float* hpb  = hp + (size_t)b * 512;

  float partial[8];
#pragma unroll
  for (int r = 0; r < 8; ++r) partial[r] = 0.f;

  for (int nt = 0; nt < 32; ++nt) {            // N = 512 units, tiles of 16
    const int n0 = nt * 16;
    // cooperative stage of W1[k=0..255][n0..n0+15] -> ldsW[n][k]
    for (int idx = tid; idx < 4096; idx += 128) {
      const int kk = idx >> 4;                 // 16 consecutive threads: one
      const int nn = idx & 15;                 // coalesced 64B row of W1
      ldsW[nn][kk] = W1[(size_t)kk * 512 + n0 + nn];
    }
    __syncthreads();

    v8f acc = {};
#pragma unroll 4
    for (int k = 0; k < 256; k += 4) {
      v2f af = *(const v2f*)(arow + k);
      v2f bf = *(const v2f*)(&ldsW[l16][k + aOff]);
      acc = wmma_f32_16x16x4(af, bf, acc);
    }
    __syncthreads();                           // tile consumed; safe to refill

    const int   ncol = n0 + l16;
    const float add  = b1[ncol] + hpb[ncol];
    const float vn   = Vw[ncol];
#pragma unroll
    for (int r = 0; r < 8; ++r)
      partial[r] += tanhf(acc[r] + add) * vn;
  }

  // deterministic cross-lane reduction within each half-wave (16 lanes = 16 units)
#pragma unroll
  for (int r = 0; r < 8; ++r) {
    float v = partial[r];
    for (int off = 8; off >= 1; off >>= 1) v += __shfl_xor(v, off, 16);
    if (l16 == 0) s_score[wave * 16 + half * 8 + r] = v + bV[0];
  }
  __syncthreads();

  // softmax over the 64 scores
  float sc = (tid < 64) ? s_score[tid] : -3.402823e38f;
  if (tid < 64) s_red[tid] = sc;
  __syncthreads();
  for (int st = 32; st >= 1; st >>= 1) {
    if (tid < st) s_red[tid] = fmaxf(s_red[tid], s_red[tid + st]);
    __syncthreads();
  }
  const float mx = s_red[0];
  __syncthreads();
  const float e = (tid < 64) ? expf(sc - mx) : 0.f;
  if (tid < 64) s_red[tid] = e;
  __syncthreads();
  for (int st = 32; st >= 1; st >>= 1) {
    if (tid < st) s_red[tid] += s_red[tid + st];
    __syncthreads();
  }
  const float inv = 1.f / s_red[0];
  if (tid < 64) {
    const float w = e * inv;
    s_w[tid] = w;
    wout[b * 64 + tid] = w;
  }
  __syncthreads();

  // context[b] = sum_s w[s] * qs[b,s,:]
  for (int em = tid; em < 256; em += 128) {
    float acc = 0.f;
    for (int s = 0; s < 64; ++s) acc += s_w[s] * Aq[(size_t)s * 256 + em];
    ctx[b * 256 + em] = acc;
  }
}

// --- small glue kernels ----------------------------------------------------
__global__ void build_gx(const int* __restrict__ x, const float* __restrict__ E,
                         const float* __restrict__ ctx, float* __restrict__ gx) {
  int i = blockIdx.x * blockDim.x + threadIdx.x;      // 64*512
  if (i >= 64 * 512) return;
  int b = i >> 9, j = i & 511;
  gx[i] = (j < 256) ? E[(size_t)x[b] * 256 + j] : ctx[b * 256 + (j - 256)];
}

// GRU step with h0 == 0: mh = bg[1] broadcast; state = (1-z)*hcand
__global__ void gru_pw(const float* __restrict__ mx, const float* __restrict__ bg1,
                       float* __restrict__ st_ws, float* __restrict__ st_out) {
  int i = blockIdx.x * blockDim.x + threadIdx.x;      // 64*256
  if (i >= 64 * 256) return;
  int b = i >> 8, g = i & 255;
  const float* m = mx + (size_t)b * 768;
  float z  = 1.f / (1.f + expf(-(m[g]       + bg1[g])));
  float r  = 1.f / (1.f + expf(-(m[256 + g] + bg1[256 + g])));
  float hc = tanhf(m[512 + g] + r * bg1[512 + g]);
  float st = (1.f - z) * hc;
  st_ws[i]  = st;
  st_out[i] = st;
}

__global__ void build_cat(const float* __restrict__ st, const float* __restrict__ feat,
                          float* __restrict__ cat) {
  int i = blockIdx.x * blockDim.x + threadIdx.x;      // 64*768
  if (i >= 64 * 768) return;
  int b = i / 768, j = i - b * 768;
  cat[i] = (j < 256) ? st[b * 256 + j] : feat[b * 512 + (j - 256)];
}

// ---------------------------------------------------------------------------
extern "C" void kernel_launch(void* const* d_in, const int* in_sizes, int n_in,
                              void* d_out, int out_size, void* d_ws, size_t ws_size,
                              hipStream_t stream) {
  const int*   x        = (const int*)  d_in[0];
  const float* qs       = (const float*)d_in[1];
  const float* features = (const float*)d_in[2];
  const float* hidden   = (const float*)d_in[3];
  const float* E        = (const float*)d_in[4];
  const float* W1       = (const float*)d_in[5];
  const float* b1       = (const float*)d_in[6];
  const float* W2       = (const float*)d_in[7];
  const float* b2       = (const float*)d_in[8];
  const float* V        = (const float*)d_in[9];
  const float* bV       = (const float*)d_in[10];
  const float* Kg       = (const float*)d_in[11];
  /* d_in[12] = Ug: dead, since h0 == 0 */
  const float* bg       = (const float*)d_in[13];  // (2, 768)
  const float* Wf1      = (const float*)d_in[14];
  const float* bf1      = (const float*)d_in[15];
  const float* Wf2      = (const float*)d_in[16];
  const float* bf2      = (const float*)d_in[17];
  const float* Wo       = (const float*)d_in[18];
  const float* bo       = (const float*)d_in[19];

  float* out       = (float*)d_out;
  float* logits    = out;                      // 64*50000
  float* state_out = out + 64 * 50000;         // 64*256
  float* w_out     = state_out + 64 * 256;     // 64*64

  float* ws  = (float*)d_ws;
  float* hp  = ws; ws += 64 * 512;
  float* ctx = ws; ws += 64 * 256;
  float* gx  = ws; ws += 64 * 512;
  float* mx  = ws; ws += 64 * 768;
  float* st  = ws; ws += 64 * 256;
  float* cat = ws; ws += 64 * 768;
  float* f1  = ws; ws += 64 * 1024;
  float* f2  = ws; ws += 64 * 1024;

  auto grid = [](int N) { return ((N / 16) + 7) / 8; };  // 8 waves per block

  gemm64_wmma<512, 512, false><<<grid(512), 256, 0, stream>>>(hidden, W2, b2, hp);
  attention_kernel<<<64, 128, 0, stream>>>(qs, W1, b1, hp, V, bV, w_out, ctx);
  build_gx<<<(64 * 512) / 256, 256, 0, stream>>>(x, E, ctx, gx);
  gemm64_wmma<512, 768, false><<<grid(768), 256, 0, stream>>>(gx, Kg, bg /*bg[0]*/, mx);
  gru_pw<<<(64 * 256) / 256, 256, 0, stream>>>(mx, bg + 768 /*bg[1]*/, st, state_out);
  build_cat<<<(64 * 768) / 256, 256, 0, stream>>>(st, features, cat);
  gemm64_wmma<768, 1024, false><<<grid(1024), 256, 0, stream>>>(cat, Wf1, bf1, f1);
  gemm64_wmma<1024, 1024, false><<<grid(1024), 256, 0, stream>>>(f1, Wf2, bf2, f2);
  gemm64_wmma<1024, 50000, true><<<grid(50000), 256, 0, stream>>>(f2, Wo, bo, logits);
}